// PQCLayer_57973468561962
// MI455X (gfx1250) — compile-verified
//
#include <hip/hip_runtime.h>
#include <hip/hip_bf16.h>

// -------------------- CDNA5 WMMA plumbing --------------------
typedef float v2f __attribute__((ext_vector_type(2)));
typedef float v8f __attribute__((ext_vector_type(8)));

static __device__ __forceinline__ v8f wmma4(v2f a, v2f b, v8f c) {
    // V_WMMA_F32_16X16X4_F32 : D(16x16 f32) = A(16x4) * B(4x16) + C
    return __builtin_amdgcn_wmma_f32_16x16x4_f32(
        /*neg_a=*/false, a, /*neg_b=*/false, b,
        /*c_mod=*/(short)0, c, /*reuse_a=*/false, /*reuse_b=*/false);
}

struct Map16 { int p[16]; };

#define N_QUBITS 16
#define N_LAYERS 4
#define BATCH    256
#define DIM      65536      // 2^16
#define CHUNK    4096       // 2^12 : qubits 4..15

// -------------------- gate matrix precompute --------------------
// For each (layer, group of 4 qubits) build U = kron(G_q0,G_q1,G_q2,G_q3),
// G_q = RZ * RY * RX (reference applies RX, then RY, then RZ).
// ws layout: float2 U[(layer*4+group)*256 + r*16 + c]
__global__ void build_gates_kernel(const float* __restrict__ params,
                                   float2* __restrict__ U) {
    int tid = blockIdx.x * blockDim.x + threadIdx.x;   // 4096 threads
    int combo = tid >> 8;                              // (layer,group) 0..15
    int layer = combo >> 2;
    int g     = combo & 3;
    int e     = tid & 255;
    int r = e >> 4, c = e & 15;

    float ar = 1.f, ai = 0.f;
    for (int t = 0; t < 4; ++t) {
        int q = 4 * g + t;
        const float* p = params + (layer * N_QUBITS + q) * 3;
        float tx = p[0] * 0.5f, ty = p[1] * 0.5f, tz = p[2] * 0.5f;
        float cx = __cosf(tx), sx = __sinf(tx);
        float cy = __cosf(ty), sy = __sinf(ty);
        float cz = __cosf(tz), sz = __sinf(tz);
        int rt = (r >> (3 - t)) & 1;   // qubit 4g is MSB of group index
        int ct = (c >> (3 - t)) & 1;
        // M = RY * RX ; RX = [[cx, -i sx],[-i sx, cx]], RY = [[cy,-sy],[sy,cy]]
        float RyM[2][2] = {{cy, -sy}, {sy, cy}};
        float RxR[2][2] = {{cx, 0.f}, {0.f, cx}};
        float RxI[2][2] = {{0.f, -sx}, {-sx, 0.f}};
        float mr = RyM[rt][0] * RxR[0][ct] + RyM[rt][1] * RxR[1][ct];
        float mi = RyM[rt][0] * RxI[0][ct] + RyM[rt][1] * RxI[1][ct];
        // RZ = diag(cz - i sz, cz + i sz)
        float er = cz, ei = (rt == 0) ? -sz : sz;
        float gr = er * mr - ei * mi;
        float gi = er * mi + ei * mr;
        float nr = ar * gr - ai * gi;
        float ni = ar * gi + ai * gr;
        ar = nr; ai = ni;
    }
    U[combo * 256 + e] = make_float2(ar, ai);
}

// -------------------- Phase B: gate on qubits 0..3 (chunk axis) --------------------
// State viewed as [B, 16(chunks), 4096]; apply 16x16 complex U0 across chunks.
// One wave per 16-wide j tile. mp maps logical chunk -> physical chunk slot.
template <bool INTERLEAVED>
__global__ void __launch_bounds__(256)
phaseB_kernel(const float* __restrict__ srcRe, const float* __restrict__ srcIm,
              const float2* __restrict__ srcC, float2* __restrict__ outC,
              const float2* __restrict__ U, Map16 mp) {
    int lane = threadIdx.x & 31;
    int wave = threadIdx.x >> 5;
    int gw = blockIdx.x * 8 + wave;     // 65536 waves total
    int batch = gw >> 8;
    int j0 = (gw & 255) * 16;
    int half = (lane < 16) ? 0 : 2;
    int lidx = lane & 15;

    // A fragments: A[m][k] = U[m][k]
    v2f Are[4], Aim[4];
    for (int kk = 0; kk < 4; ++kk) {
        int k = kk * 4 + half;
        float2 u0 = U[lidx * 16 + k];
        float2 u1 = U[lidx * 16 + k + 1];
        Are[kk] = (v2f){u0.x, u1.x};
        Aim[kk] = (v2f){u0.y, u1.y};
    }
    // B fragments: B[k][n] = state(batch, logical chunk k, j0+n)
    v2f Bre[4], Bim[4];
    size_t base = (size_t)batch * DIM + (size_t)(j0 + lidx);
    for (int kk = 0; kk < 4; ++kk) {
        int k = kk * 4 + half;
        size_t a0 = base + (size_t)mp.p[k] * CHUNK;
        size_t a1 = base + (size_t)mp.p[k + 1] * CHUNK;
        if (INTERLEAVED) {
            float2 v0 = srcC[a0], v1 = srcC[a1];
            Bre[kk] = (v2f){v0.x, v1.x};
            Bim[kk] = (v2f){v0.y, v1.y};
        } else {
            Bre[kk] = (v2f){srcRe[a0], srcRe[a1]};
            Bim[kk] = (v2f){srcIm[a0], srcIm[a1]};
        }
    }
    v8f aR = {}, aI = {};
    for (int kk = 0; kk < 4; ++kk) {
        aR = wmma4(Are[kk], Bre[kk], aR);
        aR = wmma4(-Aim[kk], Bim[kk], aR);   // Dre = Re*Re - Im*Im
        aI = wmma4(Are[kk], Bim[kk], aI);
        aI = wmma4(Aim[kk], Bre[kk], aI);
    }
    for (int r2 = 0; r2 < 8; ++r2) {
        int m = r2 + ((lane < 16) ? 0 : 8);
        size_t addr = (size_t)batch * DIM + (size_t)mp.p[m] * CHUNK
                    + (size_t)(j0 + lidx);
        outC[addr] = make_float2(aR[r2], aI[r2]);
    }
}

// -------------------- Phase A: gates on qubits 4..15 + CNOT chain --------------------
// One workgroup per (batch, chunk). Chunk of 4096 complex held interleaved in LDS.
// Bulk global<->LDS movement uses the CDNA5 async-to-LDS path (ASYNCcnt).
// Ug -> this layer's group matrices: Ug+0 = qubits 4-7, +256 = 8-11, +512 = 12-15.
__global__ void __launch_bounds__(256)
phaseA_kernel(float2* __restrict__ outC, const float2* __restrict__ Ug,
              Map16 logOfPhys) {
    __shared__ float2 lds[CHUNK];      // 32 KB interleaved (re,im)
    int tid  = threadIdx.x;
    int lane = tid & 31;
    int wave = tid >> 5;
    int pchunk = blockIdx.x & 15;
    int batch  = blockIdx.x >> 4;

    unsigned long long gptr =
        (unsigned long long)(uintptr_t)(outC + (size_t)batch * DIM
                                        + (size_t)pchunk * CHUNK);
    unsigned ldsBase = (unsigned)(uintptr_t)(&lds[0]);

    // ---- async bulk copy global -> LDS : 8 x b128 per thread (32 KB total)
    for (int j = 0; j < 8; ++j) {
        unsigned off = (unsigned)(wave * 4096 + j * 512 + lane * 16);
        asm volatile("global_load_async_to_lds_b128 %0, %1, %2"
                     :: "v"(ldsBase + off), "v"(off), "s"(gptr)
                     : "memory");
    }
    asm volatile("s_wait_asynccnt 0" ::: "memory");
    __syncthreads();

    int half = (lane < 16) ? 0 : 2;
    int lidx = lane & 15;

    // ---- group 3 (qubits 12-15, last axis): D = S * U3^T over 16 row-tiles
    {
        const float2* U3 = Ug + 2 * 256;
        for (int it = 0; it < 2; ++it) {
            int row0 = (wave * 2 + it) * 16;
            v2f Are[4], Aim[4], Bre[4], Bim[4];
            for (int kk = 0; kk < 4; ++kk) {
                int k = kk * 4 + half;
                float2 s0 = lds[(row0 + lidx) * 16 + k];
                float2 s1 = lds[(row0 + lidx) * 16 + k + 1];
                Are[kk] = (v2f){s0.x, s1.x};
                Aim[kk] = (v2f){s0.y, s1.y};
                float2 u0 = U3[lidx * 16 + k];     // B[k][n] = U3[n][k]
                float2 u1 = U3[lidx * 16 + k + 1];
                Bre[kk] = (v2f){u0.x, u1.x};
                Bim[kk] = (v2f){u0.y, u1.y};
            }
            v8f aR = {}, aI = {};
            for (int kk = 0; kk < 4; ++kk) {
                aR = wmma4(Are[kk], Bre[kk], aR);
                aR = wmma4(-Aim[kk], Bim[kk], aR);
                aI = wmma4(Are[kk], Bim[kk], aI);
                aI = wmma4(Aim[kk], Bre[kk], aI);
            }
            for (int r2 = 0; r2 < 8; ++r2) {
                int m = r2 + ((lane < 16) ? 0 : 8);
                lds[(row0 + m) * 16 + lidx] = make_float2(aR[r2], aI[r2]);
            }
        }
    }
    __syncthreads();

    // ---- group 2 (qubits 8-11, middle axis): D = U2 * S  per a-slab
    {
        const float2* U2 = Ug + 1 * 256;
        for (int it = 0; it < 2; ++it) {
            int a = wave * 2 + it;
            v2f Are[4], Aim[4], Bre[4], Bim[4];
            for (int kk = 0; kk < 4; ++kk) {
                int k = kk * 4 + half;
                float2 u0 = U2[lidx * 16 + k];     // A[m][k] = U2[m][k]
                float2 u1 = U2[lidx * 16 + k + 1];
                Are[kk] = (v2f){u0.x, u1.x};
                Aim[kk] = (v2f){u0.y, u1.y};
                float2 s0 = lds[a * 256 + k * 16 + lidx];
                float2 s1 = lds[a * 256 + (k + 1) * 16 + lidx];
                Bre[kk] = (v2f){s0.x, s1.x};
                Bim[kk] = (v2f){s0.y, s1.y};
            }
            v8f aR = {}, aI = {};
            for (int kk = 0; kk < 4; ++kk) {
                aR = wmma4(Are[kk], Bre[kk], aR);
                aR = wmma4(-Aim[kk], Bim[kk], aR);
                aI = wmma4(Are[kk], Bim[kk], aI);
                aI = wmma4(Aim[kk], Bre[kk], aI);
            }
            for (int r2 = 0; r2 < 8; ++r2) {
                int m = r2 + ((lane < 16) ? 0 : 8);
                lds[a * 256 + m * 16 + lidx] = make_float2(aR[r2], aI[r2]);
            }
        }
    }
    __syncthreads();

    // ---- group 1 (qubits 4-7, first axis): D = U1 * S per b-slab
    {
        const float2* U1 = Ug;
        for (int it = 0; it < 2; ++it) {
            int b = wave * 2 + it;
            v2f Are[4], Aim[4], Bre[4], Bim[4];
            for (int kk = 0; kk < 4; ++kk) {
                int k = kk * 4 + half;
                float2 u0 = U1[lidx * 16 + k];
                float2 u1 = U1[lidx * 16 + k + 1];
                Are[kk] = (v2f){u0.x, u1.x};
                Aim[kk] = (v2f){u0.y, u1.y};
                float2 s0 = lds[k * 256 + b * 16 + lidx];
                float2 s1 = lds[(k + 1) * 256 + b * 16 + lidx];
                Bre[kk] = (v2f){s0.x, s1.x};
                Bim[kk] = (v2f){s0.y, s1.y};
            }
            v8f aR = {}, aI = {};
            for (int kk = 0; kk < 4; ++kk) {
                aR = wmma4(Are[kk], Bre[kk], aR);
                aR = wmma4(-Aim[kk], Bim[kk], aR);
                aI = wmma4(Are[kk], Bim[kk], aI);
                aI = wmma4(Aim[kk], Bre[kk], aI);
            }
            for (int r2 = 0; r2 < 8; ++r2) {
                int m = r2 + ((lane < 16) ? 0 : 8);
                lds[m * 256 + b * 16 + lidx] = make_float2(aR[r2], aI[r2]);
            }
        }
    }
    __syncthreads();

    // ---- CNOT chain c=3..14 inside chunk.
    // control for CNOT(3,4) = parity of logical chunk bits (accounts for c=0..2).
    // chain c=4..14 => each bit becomes prefix-XOR (from MSB) of original bits.
    int l  = logOfPhys.p[pchunk];
    int cp = __popc(l) & 1;
    float2 tv[16];
    int    ty[16];
    int cnt = 0;
    for (int i = tid; i < CHUNK; i += 256) {
        int t = i;
        t ^= t >> 1; t ^= t >> 2; t ^= t >> 4; t ^= t >> 8;
        t &= 0xFFF;
        if (cp) t ^= 0xFFF;
        ty[cnt] = t; tv[cnt] = lds[i]; ++cnt;
    }
    __syncthreads();
    for (int k = 0; k < 16; ++k) lds[ty[k]] = tv[k];
    __syncthreads();

    // ---- async bulk copy LDS -> global
    for (int j = 0; j < 8; ++j) {
        unsigned off = (unsigned)(wave * 4096 + j * 512 + lane * 16);
        asm volatile("global_store_async_from_lds_b128 %0, %1, %2"
                     :: "v"(off), "v"(ldsBase + off), "s"(gptr)
                     : "memory");
    }
    asm volatile("s_wait_asynccnt 0" ::: "memory");
    // S_ENDPGM performs implicit wait-idle as well.
}

// -------------------- host side --------------------
static inline int pstep4(int l) {           // CNOTs (0,1),(1,2),(2,3) on top-4 bits
    int t = l ^ (l >> 1);
    t ^= (t >> 2);
    return t & 15;
}

extern "C" void kernel_launch(void* const* d_in, const int* in_sizes, int n_in,
                              void* d_out, int out_size, void* d_ws, size_t ws_size,
                              hipStream_t stream) {
    const float* state_re = (const float*)d_in[0];
    const float* state_im = (const float*)d_in[1];
    const float* params   = (const float*)d_in[2];
    float2* outC = (float2*)d_out;
    float2* U    = (float2*)d_ws;          // 4 layers * 4 groups * 256 entries = 32KB

    (void)in_sizes; (void)n_in; (void)out_size; (void)ws_size;

    // chunk permutation powers: P^0..P^3 (P^4 == identity over GF(2))
    int P[4][16];
    for (int i = 0; i < 16; ++i) P[0][i] = i;
    for (int k = 1; k < 4; ++k)
        for (int i = 0; i < 16; ++i) P[k][i] = pstep4(P[k - 1][i]);

    build_gates_kernel<<<16, 256, 0, stream>>>(params, U);

    for (int L = 0; L < N_LAYERS; ++L) {
        // Phase B: U0 across chunk axis. physOfLogical = P^{-L} = P^{(4-L)%4}
        Map16 mb;
        for (int i = 0; i < 16; ++i) mb.p[i] = P[(4 - L) & 3][i];
        if (L == 0) {
            phaseB_kernel<false><<<8192, 256, 0, stream>>>(
                state_re, state_im, nullptr, outC, U + (L * 4 + 0) * 256, mb);
        } else {
            phaseB_kernel<true><<<8192, 256, 0, stream>>>(
                nullptr, nullptr, (const float2*)d_out, outC,
                U + (L * 4 + 0) * 256, mb);
        }

        // Phase A: logOfPhys = P^{L}
        Map16 ma;
        for (int i = 0; i < 16; ++i) ma.p[i] = P[L][i];
        phaseA_kernel<<<BATCH * 16, 256, 0, stream>>>(outC,
                                                      U + (L * 4 + 1) * 256, ma);
    }
    // After 4 layers the cumulative chunk relabel P^4 is the identity:
    // d_out already holds [B, DIM, (re,im)] in logical order.
}